// WavelengthDependentPropagation_38208029065721
// MI455X (gfx1250) — compile-verified
//
#include <hip/hip_runtime.h>
#include <math.h>

// ---------------------------------------------------------------------------
// Wavelength-dependent angular-spectrum propagation on gfx1250 (MI455X).
//
//   pass1: forward FFT along rows   (global -> ws, interleaved complex)
//   pass2: forward FFT along cols -> multiply by H -> inverse FFT cols
//          (column panels moved HBM<->LDS by the Tensor Data Mover)
//   pass3: inverse FFT along rows   (ws -> planar real/imag output)
//
// Each 1024-point FFT = radix-4 VALU stage + two DFT-16 stages done as
// complex 16x16x16 matrix multiplies on V_WMMA_F32_16X16X4_F32 (f32 WMMA).
// One wave32 owns one 1D FFT in LDS; no intra-wave barriers required.
// ---------------------------------------------------------------------------

typedef float v2f __attribute__((ext_vector_type(2)));
typedef float v8f __attribute__((ext_vector_type(8)));
typedef unsigned int v4u __attribute__((ext_vector_type(4)));
typedef int v4i __attribute__((ext_vector_type(4)));
typedef int v8i __attribute__((ext_vector_type(8)));

#define PI2F 6.28318530717958647692f
#define FREQ_SCALE 122.0703125f   // 1 / (1024 * 8e-6)
#define ZDIST 0.05f
#define NPIX 1024
#define NIMG 24                   // 8 batch * 3 wavelengths

static __device__ __forceinline__ float2 cmulf(float2 a, float2 b) {
  return make_float2(a.x * b.x - a.y * b.y, a.x * b.y + a.y * b.x);
}

// ---- f32 WMMA 16x16x4 wrapper (A,B = 2 VGPRs each; C/D = 8 VGPRs) ---------
static __device__ __forceinline__ v8f wmma4(v2f a, v2f b, v8f c) {
  return __builtin_amdgcn_wmma_f32_16x16x4_f32(false, a, false, b, (short)0, c,
                                               false, false);
}

// ---- Tensor Data Mover: 1024x4 complex64 tile <-> LDS ---------------------
// D# built per CDNA5 ISA 8.3/8.4:
//   group0: count=1 | lds_addr | global_addr[56:0] | type=2
//   group1: data_size=3 (8B), tensor_dim0=1024, tensor_dim1=1024,
//           tile_dim0=4, tile_dim1=1024, tensor_dim0_stride=1024
// 2D tensor -> higher-dim groups zero (dims 2+ unused).
// This toolchain exposes the 6-arg builtin form:
//   (v4u g0, v8i g1, v4i, v4i, v8i, i32 cpol)
static __device__ __forceinline__ void tdm_panel(int store, unsigned lds_addr,
                                                 unsigned long long gaddr) {
  v4u g0;
  g0[0] = 1u;                                       // count=1 user descriptor
  g0[1] = lds_addr;                                 // LDS byte address
  g0[2] = (unsigned)gaddr;                          // global_addr[31:0]
  g0[3] = (unsigned)(gaddr >> 32) | 0x80000000u;    // addr[56:32] | type=2<<30
  v8i g1;
  g1[0] = 0x00030000;   // wg_mask=0, data_size=3 (8B), no pad/iterate
  g1[1] = 0x04000000;   // atomic_barrier_addr=0 | tensor_dim0 lo16 = 1024
  g1[2] = 0x04000000;   // tensor_dim0 hi16=0   | tensor_dim1 lo16 = 1024
  g1[3] = 0x00040000;   // tensor_dim1 hi16=0   | tile_dim0 = 4
  g1[4] = 0x00000400;   // tile_dim1 = 1024, tile_dim2 = 0
  g1[5] = 1024;         // tensor_dim0_stride lo32 = 1024 elements
  g1[6] = 0;            // stride hi16 | tensor_dim1_stride lo16
  g1[7] = 0;            // tensor_dim1_stride hi32
  v4i gz4 = {0, 0, 0, 0};
  v8i gz8 = {0, 0, 0, 0, 0, 0, 0, 0};
  if (store)
    __builtin_amdgcn_tensor_store_from_lds(g0, g1, gz4, gz4, gz8, 0);
  else
    __builtin_amdgcn_tensor_load_to_lds(g0, g1, gz4, gz4, gz8, 0);
}

// Build DFT16 matrix F[m][k] = exp(dir * 2*pi*i * m*k / 16) in the WMMA
// A-operand register layout:
//   chunk c (K = 4c..4c+3), component j: lane<16 -> (M=lane,   K=4c+j)
//                                        lane>=16 -> (M=lane-16, K=4c+2+j)
static __device__ __forceinline__ void build_dft16(float dir, v2f* Fr, v2f* Fi,
                                                   v2f* FiN) {
  int lane = threadIdx.x & 31;
  int m = lane & 15, hi = lane >> 4;
#pragma unroll
  for (int c = 0; c < 4; ++c) {
    v2f r, im, imn;
#pragma unroll
    for (int j = 0; j < 2; ++j) {
      int k = 4 * c + 2 * hi + j;
      float ang = dir * (PI2F / 16.f) * (float)((m * k) & 15);
      float s, cc;
      __sincosf(ang, &s, &cc);
      r[j] = cc; im[j] = s; imn[j] = -s;
    }
    Fr[c] = r; Fi[c] = im; FiN[c] = imn;
  }
}

// Complex 16x16 = F(16x16) * B(16x16), B gathered from LDS:
//   B[k][col] = lds[base + k*ks + col*cs]
// Dr = Fr*Br + FiN*Bi ; Di = Fr*Bi + Fi*Br   (16 wmma ops)
static __device__ __forceinline__ void wmma_c16(const v2f* Fr, const v2f* Fi,
                                                const v2f* FiN,
                                                const float2* lds, int base,
                                                int ks, int cs, v8f& dR,
                                                v8f& dI) {
  int lane = threadIdx.x & 31;
  int hi = lane >> 4, col = lane & 15;
  v8f z0 = {0.f, 0.f, 0.f, 0.f, 0.f, 0.f, 0.f, 0.f};
  dR = z0; dI = z0;
#pragma unroll
  for (int c = 0; c < 4; ++c) {
    v2f br, bi;
#pragma unroll
    for (int j = 0; j < 2; ++j) {
      int k = 4 * c + 2 * hi + j;
      float2 z = lds[base + k * ks + col * cs];
      br[j] = z.x; bi[j] = z.y;
    }
    dR = wmma4(Fr[c], br, dR);
    dR = wmma4(FiN[c], bi, dR);
    dI = wmma4(Fr[c], bi, dI);
    dI = wmma4(Fi[c], br, dI);
  }
}

// Radix-4 butterfly + W64^{m1*j2} twiddle, scatter into U layout
// U[(j2*256 + m1*16 + n1) * so]
static __device__ __forceinline__ void radix4_store(float2 a0, float2 a1,
                                                    float2 a2, float2 a3,
                                                    int m1, int n1, float2* out,
                                                    int so, float dir) {
  const float TW = dir * (PI2F / 64.f);
  float2 e0 = make_float2(a0.x + a2.x, a0.y + a2.y);
  float2 e1 = make_float2(a0.x - a2.x, a0.y - a2.y);
  float2 o0 = make_float2(a1.x + a3.x, a1.y + a3.y);
  float2 o1 = make_float2(a1.x - a3.x, a1.y - a3.y);
  float2 jo = make_float2(-dir * o1.y, dir * o1.x);  // dir * i * o1
  float2 b0 = make_float2(e0.x + o0.x, e0.y + o0.y);
  float2 b2 = make_float2(e0.x - o0.x, e0.y - o0.y);
  float2 b1 = make_float2(e1.x + jo.x, e1.y + jo.y);
  float2 b3 = make_float2(e1.x - jo.x, e1.y - jo.y);
  int ub = m1 * 16 + n1;
  out[ub * so] = b0;
  float s, c;
  __sincosf(TW * (float)m1, &s, &c);
  out[(256 + ub) * so] = cmulf(b1, make_float2(c, s));
  __sincosf(TW * (float)(2 * m1), &s, &c);
  out[(512 + ub) * so] = cmulf(b2, make_float2(c, s));
  __sincosf(TW * (float)(3 * m1), &s, &c);
  out[(768 + ub) * so] = cmulf(b3, make_float2(c, s));
}

// Stage 0: 256 radix-4 FFTs over x[n1 + 16*m1 + 256*m2]
static __device__ __forceinline__ void stage0(const float2* in, int si,
                                              float2* out, int so, float dir) {
  int lane = threadIdx.x & 31;
#pragma unroll
  for (int it = 0; it < 8; ++it) {
    int p = it * 32 + lane;
    int n1 = p & 15, m1 = p >> 4;
    int b = n1 + 16 * m1;
    float2 a0 = in[b * si];
    float2 a1 = in[(b + 256) * si];
    float2 a2 = in[(b + 512) * si];
    float2 a3 = in[(b + 768) * si];
    radix4_store(a0, a1, a2, a3, m1, n1, out, so, dir);
  }
}

// Stage 0 reading from two planar f32 arrays (pass 1 input)
static __device__ __forceinline__ void stage0_planar(const float* xr,
                                                     const float* xi,
                                                     float2* out, float dir) {
  int lane = threadIdx.x & 31;
#pragma unroll
  for (int it = 0; it < 8; ++it) {
    int p = it * 32 + lane;
    int n1 = p & 15, m1 = p >> 4;
    int b = n1 + 16 * m1;
    float2 a0 = make_float2(xr[b], xi[b]);
    float2 a1 = make_float2(xr[b + 256], xi[b + 256]);
    float2 a2 = make_float2(xr[b + 512], xi[b + 512]);
    float2 a3 = make_float2(xr[b + 768], xi[b + 768]);
    radix4_store(a0, a1, a2, a3, m1, n1, out, 1, dir);
  }
}

// Stage 1: DFT16 over m1 (WMMA) + outer twiddle W1024^{n1*k2};
// writes V[(k2*16 + n1) * so] with k2 = t + 4*j1
static __device__ __forceinline__ void stage1(const float2* U, int si,
                                              float2* out, int so, float dir,
                                              const v2f* Fr, const v2f* Fi,
                                              const v2f* FiN) {
  int lane = threadIdx.x & 31;
  int hi = lane >> 4, col = lane & 15;
  const float TW = dir * (PI2F / 1024.f);
#pragma unroll
  for (int t = 0; t < 4; ++t) {
    v8f dR, dI;
    wmma_c16(Fr, Fi, FiN, U, t * 256 * si, 16 * si, si, dR, dI);
#pragma unroll
    for (int r = 0; r < 8; ++r) {
      int j1 = r + 8 * hi;
      int n1 = col;
      int k2 = t + 4 * j1;
      float s, c;
      __sincosf(TW * (float)(n1 * k2), &s, &c);
      float xr = dR[r], xi = dI[r];
      out[(k2 * 16 + n1) * so] = make_float2(xr * c - xi * s, xr * s + xi * c);
    }
  }
}

// ---------------------------------------------------------------------------
// Pass 1: forward FFT along rows.  4 waves / block, one row per wave.
// ---------------------------------------------------------------------------
__global__ void __launch_bounds__(128) k_fft_rows_fwd(
    const float* __restrict__ xr, const float* __restrict__ xi,
    float2* __restrict__ ws) {
  __shared__ float2 Ub[4][1024];
  __shared__ float2 Vb[4][1024];
  int wave = threadIdx.x >> 5;
  int lane = threadIdx.x & 31;
  int hi = lane >> 4, col = lane & 15;
  long rowg = (long)blockIdx.x * 4 + wave;
  long base = rowg * NPIX;

  v2f Fr[4], Fi[4], FiN[4];
  build_dft16(-1.f, Fr, Fi, FiN);

  stage0_planar(xr + base, xi + base, Ub[wave], -1.f);
  stage1(Ub[wave], 1, Vb[wave], 1, -1.f, Fr, Fi, FiN);

  // Stage 2: DFT16 over n1, write X[64*k1 + k2] straight to global
#pragma unroll
  for (int u = 0; u < 4; ++u) {
    v8f dR, dI;
    wmma_c16(Fr, Fi, FiN, Vb[wave], 256 * u, 1, 16, dR, dI);
#pragma unroll
    for (int r = 0; r < 8; ++r) {
      int k1 = r + 8 * hi;
      int idx = 64 * k1 + 16 * u + col;
      ws[base + idx] = make_float2(dR[r], dI[r]);
    }
  }
}

// ---------------------------------------------------------------------------
// Pass 2: forward FFT along columns -> multiply by H(lambda,fy,fx) ->
//         inverse FFT along columns (scaled 1/1024).  4 columns per block;
//         panels moved by the Tensor Data Mover.
// ---------------------------------------------------------------------------
__global__ void __launch_bounds__(128) k_fft_cols_h(
    float2* __restrict__ ws, const float* __restrict__ wl) {
  __shared__ float2 panel[1024 * 4];  // [row][col-in-block], TDM tile layout
  __shared__ float2 Ub[4][1024];
  int t = threadIdx.x;
  int img = blockIdx.x >> 8;        // 256 column-groups per image
  int cg = blockIdx.x & 255;
  int col0 = cg * 4;
  long ibase = (long)img * NPIX * NPIX;

  unsigned panel_lds = (unsigned)(unsigned long long)(uintptr_t)panel;
  unsigned long long tile_gaddr =
      (unsigned long long)(uintptr_t)(ws + ibase + col0);

  // one wave issues the DMA (TDM ignores EXEC, so use a scalarized branch)
  if (__builtin_amdgcn_readfirstlane(threadIdx.x >> 5) == 0) {
    tdm_panel(0, panel_lds, tile_gaddr);
    __builtin_amdgcn_s_wait_tensorcnt(0);
  }
  __syncthreads();

  int wave = t >> 5;
  int lane = t & 31;
  int hi = lane >> 4, col = lane & 15;
  float2* P = panel + wave;  // element n of this wave's column at P[n*4]
  float2* Uw = Ub[wave];

  v2f Fr[4], Fi[4], FiN[4];
  build_dft16(-1.f, Fr, Fi, FiN);

  // ---- forward column FFT ----
  stage0(P, 4, Uw, 1, -1.f);
  stage1(Uw, 1, P, 4, -1.f, Fr, Fi, FiN);

  // stage 2 + transfer function H, result (natural order) into Uw
  int colg = col0 + wave;
  int ch = img % 3;
  float lam = wl[ch];
  float lam2 = lam * lam;
  float kzc = PI2F * ZDIST / lam;
  float fxv = (float)(colg < 512 ? colg : colg - 1024) * FREQ_SCALE;
  float fx2 = fxv * fxv;
#pragma unroll
  for (int u = 0; u < 4; ++u) {
    v8f dR, dI;
    wmma_c16(Fr, Fi, FiN, P, 256 * u * 4, 4, 64, dR, dI);
#pragma unroll
    for (int r = 0; r < 8; ++r) {
      int k1 = r + 8 * hi;
      int rf = 64 * k1 + 16 * u + col;  // frequency row index
      float fyv = (float)(rf < 512 ? rf : rf - 1024) * FREQ_SCALE;
      float arg = 1.f - lam2 * (fx2 + fyv * fyv);
      float2 h = make_float2(0.f, 0.f);
      if (arg > 0.f) {
        float s, c;
        __sincosf(kzc * __builtin_sqrtf(arg), &s, &c);
        h = make_float2(c, s);
      }
      Uw[rf] = cmulf(make_float2(dR[r], dI[r]), h);
    }
  }

  // ---- inverse column FFT (conjugate twiddles: dir=+1, Fi<->FiN) ----
  stage0(Uw, 1, P, 4, 1.f);
  stage1(P, 4, Uw, 1, 1.f, Fr, FiN, Fi);
  const float S = 1.f / 1024.f;
#pragma unroll
  for (int u = 0; u < 4; ++u) {
    v8f dR, dI;
    wmma_c16(Fr, FiN, Fi, Uw, 256 * u, 1, 16, dR, dI);
#pragma unroll
    for (int r = 0; r < 8; ++r) {
      int k1 = r + 8 * hi;
      int idx = 64 * k1 + 16 * u + col;
      P[idx * 4] = make_float2(dR[r] * S, dI[r] * S);
    }
  }
  __syncthreads();

  // DMA the panel back (kernel-end implicit wait-idle drains TENSORcnt)
  if (__builtin_amdgcn_readfirstlane(threadIdx.x >> 5) == 0) {
    tdm_panel(1, panel_lds, tile_gaddr);
    __builtin_amdgcn_s_wait_tensorcnt(0);
  }
}

// ---------------------------------------------------------------------------
// Pass 3: inverse FFT along rows, scale 1/1024, write planar real/imag.
// ---------------------------------------------------------------------------
__global__ void __launch_bounds__(128) k_ifft_rows(
    const float2* __restrict__ ws, float* __restrict__ outr,
    float* __restrict__ outi) {
  __shared__ float2 Ub[4][1024];
  __shared__ float2 Vb[4][1024];
  int wave = threadIdx.x >> 5;
  int lane = threadIdx.x & 31;
  int hi = lane >> 4, col = lane & 15;
  long rowg = (long)blockIdx.x * 4 + wave;
  long base = rowg * NPIX;

  v2f Fr[4], Fi[4], FiN[4];
  build_dft16(-1.f, Fr, Fi, FiN);  // inverse uses Fi<->FiN swap + dir=+1

  stage0(ws + base, 1, Ub[wave], 1, 1.f);
  stage1(Ub[wave], 1, Vb[wave], 1, 1.f, Fr, FiN, Fi);

  const float S = 1.f / 1024.f;
#pragma unroll
  for (int u = 0; u < 4; ++u) {
    v8f dR, dI;
    wmma_c16(Fr, FiN, Fi, Vb[wave], 256 * u, 1, 16, dR, dI);
#pragma unroll
    for (int r = 0; r < 8; ++r) {
      int k1 = r + 8 * hi;
      long idx = base + 64 * k1 + 16 * u + col;
      outr[idx] = dR[r] * S;
      outi[idx] = dI[r] * S;
    }
  }
}

// ---------------------------------------------------------------------------
extern "C" void kernel_launch(void* const* d_in, const int* in_sizes, int n_in,
                              void* d_out, int out_size, void* d_ws,
                              size_t ws_size, hipStream_t stream) {
  const float* xr = (const float*)d_in[0];
  const float* xi = (const float*)d_in[1];
  const float* wl = (const float*)d_in[2];
  float2* ws = (float2*)d_ws;  // 24 * 1024 * 1024 complex64 = 192 MB
  float* outr = (float*)d_out;
  float* outi = outr + (size_t)NIMG * NPIX * NPIX;

  dim3 blk(128);
  dim3 grd(6144);  // 24576 rows / 4 per block ; 24 imgs * 256 col-groups
  k_fft_rows_fwd<<<grd, blk, 0, stream>>>(xr, xi, ws);
  k_fft_cols_h<<<grd, blk, 0, stream>>>(ws, wl);
  k_ifft_rows<<<grd, blk, 0, stream>>>(ws, outr, outi);
}